// MvdrBeamformer_74869869904734
// MI455X (gfx1250) — compile-verified
//
#include <hip/hip_runtime.h>
#include <math.h>

#define N_DIM 8
#define C_DIM 8
#define F_DIM 257
#define T_DIM 1000
#define NF (N_DIM * F_DIM)
#define EPSF 1.1920929e-07f   // float32 machine epsilon
#define EPS5 1e-5f

typedef float v2f __attribute__((ext_vector_type(2)));
typedef float v8f __attribute__((ext_vector_type(8)));

// ---------------------------------------------------------------------------
// Kernel 1: masked spatial covariance via V_WMMA_F32_16X16X4_F32.
// One block per (n,f). 4 waves, each accumulates a 16x16 f32 WMMA tile over a
// strided set of 4-frame chunks. A rows 0-7 = Xr channels, rows 8-15 = Xi
// channels; B columns likewise (mask-weighted for acc_s, raw for acc_t).
// D[i][j](i,j<8)=P_rr, D[i][j+8]=P_ri, D[i+8][j]=P_ir, D[i+8][j+8]=P_ii.
// Re(nom)=P_rr+P_ii, Im(nom)=P_ir-P_ri. Rn_nom = total_nom - masked_nom.
// The normalized mask row is staged in LDS; `wave` is made wave-uniform via
// readfirstlane so the K-loop is a scalar loop with EXEC = all ones (WMMA
// requirement) instead of an exec-masked divergent loop.
// ---------------------------------------------------------------------------
__global__ __launch_bounds__(128) void covar_kernel(
    const float* __restrict__ mask, const float* __restrict__ xr,
    const float* __restrict__ xi, float* __restrict__ Rs_out,
    float* __restrict__ Rn_out, float* __restrict__ feat_out) {
  const int nf = blockIdx.x;
  const int n = nf / F_DIM, f = nf - n * F_DIM;
  const int tid = threadIdx.x;
  const int wave = __builtin_amdgcn_readfirstlane(tid) >> 5;  // uniform SGPR
  const int lane = tid & 31;

  __shared__ __align__(16) float smask[T_DIM];  // normalized mask row
  __shared__ float red_max[128];
  __shared__ float red_sum[128];

  // ---- pass 1: stage mask row to LDS; reduce max|m| and sum(m) ----
  const float* mrow = mask + (size_t)n * T_DIM * F_DIM + f;
  float mmax = 0.f, msum = 0.f;
  for (int t = tid; t < T_DIM; t += 128) {
    float m = mrow[(size_t)t * F_DIM];
    smask[t] = m;
    mmax = fmaxf(mmax, fabsf(m));
    msum += m;
  }
  red_max[tid] = mmax;
  red_sum[tid] = msum;
  __syncthreads();
  for (int s = 64; s > 0; s >>= 1) {
    if (tid < s) {
      red_max[tid] = fmaxf(red_max[tid], red_max[tid + s]);
      red_sum[tid] += red_sum[tid + s];
    }
    __syncthreads();
  }
  const float inv_max = 1.f / (red_max[0] + EPSF);
  const float sum_ms = red_sum[0] * inv_max;
  const float inv_den_s = 1.f / fmaxf(sum_ms, EPSF);
  const float inv_den_n = 1.f / fmaxf((float)T_DIM - sum_ms, EPSF);
  // normalize the staged mask in place (each thread rescales what it wrote)
  for (int t = tid; t < T_DIM; t += 128) smask[t] *= inv_max;
  __syncthreads();

  // ---- pass 2: WMMA accumulation over T ----
  const int part = (lane >> 3) & 1;      // row/col 0-7 -> real, 8-15 -> imag
  const int ch = lane & 7;               // channel
  const int koff = (lane >> 4) << 1;     // lanes 16-31 hold K=2,3
  const float* xp = part ? xi : xr;
  const float* xrow =
      xp + (((size_t)n * C_DIM + ch) * F_DIM + f) * (size_t)T_DIM;

  v8f acc_s = {};  // masked (ms-weighted) products
  v8f acc_t = {};  // unweighted (total) products
  for (int chunk = wave; chunk < T_DIM / 4; chunk += 4) {
    const int t = chunk * 4 + koff;
    __builtin_prefetch(xrow + t + 16, 0, 0);  // next chunk for this wave
    v2f a = *(const v2f*)(xrow + t);          // t even -> 8B aligned
    v2f m = *(const v2f*)(&smask[t]);         // ds_load_b64 (broadcast)
    v2f b;
    b[0] = a[0] * m[0];
    b[1] = a[1] * m[1];
    acc_t = __builtin_amdgcn_wmma_f32_16x16x4_f32(false, a, false, a,
                                                  (short)0, acc_t, false, false);
    acc_s = __builtin_amdgcn_wmma_f32_16x16x4_f32(false, a, false, b,
                                                  (short)0, acc_s, false, false);
  }

  // ---- combine the 4 per-wave tiles and assemble complex matrices ----
  __shared__ float lds_s[4][8][32];
  __shared__ float lds_t[4][8][32];
#pragma unroll
  for (int r = 0; r < 8; ++r) {
    lds_s[wave][r][lane] = acc_s[r];
    lds_t[wave][r][lane] = acc_t[r];
  }
  __syncthreads();

  if (tid < 64) {
    const int i = tid >> 3, j = tid & 7;
    float s_rr = 0, s_ri = 0, s_ir = 0, s_ii = 0;
    float t_rr = 0, t_ri = 0, t_ir = 0, t_ii = 0;
#pragma unroll
    for (int w = 0; w < 4; ++w) {
      s_rr += lds_s[w][i][j];
      s_ri += lds_s[w][i][j + 8];
      s_ir += lds_s[w][i][16 + j];
      s_ii += lds_s[w][i][24 + j];
      t_rr += lds_t[w][i][j];
      t_ri += lds_t[w][i][j + 8];
      t_ir += lds_t[w][i][16 + j];
      t_ii += lds_t[w][i][24 + j];
    }
    const float nsr = s_rr + s_ii, nsi = s_ir - s_ri;
    const float ntr = t_rr + t_ii, nti = t_ir - t_ri;
    const float Rs_re = nsr * inv_den_s, Rs_im = nsi * inv_den_s;
    float Rn_re = (ntr - nsr) * inv_den_n;
    const float Rn_im = (nti - nsi) * inv_den_n;
    if (i == j) Rn_re += EPS5;  // Rn + eps*I
    const size_t base = ((size_t)nf * 64 + tid) * 2;
    Rs_out[base] = Rs_re;
    Rs_out[base + 1] = Rs_im;
    Rn_out[base] = Rn_re;
    Rn_out[base + 1] = Rn_im;
  }
  __syncthreads();

  // feat[n,c,f] = | sum_{j!=i} Rs[i][j] / (C-1) |
  if (tid < 8) {
    const int i = tid;
    float sre = 0.f, sim = 0.f;
    for (int j = 0; j < 8; ++j) {
      if (j == i) continue;
      float s_rr = 0, s_ri = 0, s_ir = 0, s_ii = 0;
#pragma unroll
      for (int w = 0; w < 4; ++w) {
        s_rr += lds_s[w][i][j];
        s_ri += lds_s[w][i][j + 8];
        s_ir += lds_s[w][i][16 + j];
        s_ii += lds_s[w][i][24 + j];
      }
      sre += s_rr + s_ii;
      sim += s_ir - s_ri;
    }
    sre *= inv_den_s * (1.f / 7.f);
    sim *= inv_den_s * (1.f / 7.f);
    feat_out[((size_t)n * C_DIM + i) * F_DIM + f] = sqrtf(sre * sre + sim * sim);
  }
}

// ---------------------------------------------------------------------------
// Kernel 2: g[n,c] = gvec_w . tanh(feat[n,c,:] @ proj_w^T + proj_b) + gvec_b
// One block per (n,c); 256 threads each own 2 of the 512 att dims.
// ---------------------------------------------------------------------------
__global__ __launch_bounds__(256) void attn_kernel(
    const float* __restrict__ feat, const float* __restrict__ proj_w,
    const float* __restrict__ proj_b, const float* __restrict__ gvec_w,
    const float* __restrict__ gvec_b, float* __restrict__ g_out) {
  const int nc = blockIdx.x;
  const float* frow = feat + (size_t)nc * F_DIM;
  __shared__ float sfeat[F_DIM];
  for (int f = threadIdx.x; f < F_DIM; f += 256) sfeat[f] = frow[f];
  __syncthreads();

  float part = 0.f;
  for (int d = threadIdx.x; d < 512; d += 256) {
    const float* wrow = proj_w + (size_t)d * F_DIM;
    float acc = proj_b[d];
    for (int f = 0; f < F_DIM; ++f) acc = fmaf(sfeat[f], wrow[f], acc);
    part += tanhf(acc) * gvec_w[d];
  }
  __shared__ float red[256];
  red[threadIdx.x] = part;
  __syncthreads();
  for (int s = 128; s > 0; s >>= 1) {
    if (threadIdx.x < s) red[threadIdx.x] += red[threadIdx.x + s];
    __syncthreads();
  }
  if (threadIdx.x == 0) g_out[nc] = red[0] + gvec_b[0];
}

// ---------------------------------------------------------------------------
// Kernel 3: per-(n,f) MVDR solve. softmax(g) -> u; Gauss-Jordan inverse of
// Rn (already +eps*I, Hermitian PD so no pivoting needed); w = RiRs@u/(tr+eps);
// store conj(w).
// ---------------------------------------------------------------------------
__global__ void solve_kernel(const float* __restrict__ Rs_in,
                             const float* __restrict__ Rn_in,
                             const float* __restrict__ g_in,
                             float* __restrict__ wv_out) {
  const int nf = blockIdx.x * blockDim.x + threadIdx.x;
  if (nf >= NF) return;
  const int n = nf / F_DIM;

  // softmax over channels
  float u[8];
  float gmax = -1e30f;
  for (int c = 0; c < 8; ++c) {
    u[c] = g_in[n * 8 + c];
    gmax = fmaxf(gmax, u[c]);
  }
  float se = 0.f;
  for (int c = 0; c < 8; ++c) {
    u[c] = expf(u[c] - gmax);
    se += u[c];
  }
  const float ise = 1.f / se;
  for (int c = 0; c < 8; ++c) u[c] *= ise;

  float A[8][8][2], B[8][8][2], S[8][8][2];
  const float* rn = Rn_in + (size_t)nf * 128;
  const float* rs = Rs_in + (size_t)nf * 128;
  for (int i = 0; i < 8; ++i)
    for (int j = 0; j < 8; ++j) {
      A[i][j][0] = rn[(i * 8 + j) * 2];
      A[i][j][1] = rn[(i * 8 + j) * 2 + 1];
      S[i][j][0] = rs[(i * 8 + j) * 2];
      S[i][j][1] = rs[(i * 8 + j) * 2 + 1];
      B[i][j][0] = (i == j) ? 1.f : 0.f;
      B[i][j][1] = 0.f;
    }

  // complex Gauss-Jordan (no pivoting)
  for (int k = 0; k < 8; ++k) {
    const float pr = A[k][k][0], pi = A[k][k][1];
    const float d = 1.f / (pr * pr + pi * pi);
    const float ir = pr * d, ii = -pi * d;
    for (int j = 0; j < 8; ++j) {
      float ar = A[k][j][0], ai = A[k][j][1];
      A[k][j][0] = ar * ir - ai * ii;
      A[k][j][1] = ar * ii + ai * ir;
      float br = B[k][j][0], bi = B[k][j][1];
      B[k][j][0] = br * ir - bi * ii;
      B[k][j][1] = br * ii + bi * ir;
    }
    for (int r = 0; r < 8; ++r) {
      if (r == k) continue;
      const float fr = A[r][k][0], fi = A[r][k][1];
      for (int j = 0; j < 8; ++j) {
        A[r][j][0] -= fr * A[k][j][0] - fi * A[k][j][1];
        A[r][j][1] -= fr * A[k][j][1] + fi * A[k][j][0];
        B[r][j][0] -= fr * B[k][j][0] - fi * B[k][j][1];
        B[r][j][1] -= fr * B[k][j][1] + fi * B[k][j][0];
      }
    }
  }

  // s[j] = sum_c Rs[j][c] * u[c]   (u real)
  float sv[8][2];
  for (int j = 0; j < 8; ++j) {
    float ar = 0.f, ai = 0.f;
    for (int c = 0; c < 8; ++c) {
      ar += S[j][c][0] * u[c];
      ai += S[j][c][1] * u[c];
    }
    sv[j][0] = ar;
    sv[j][1] = ai;
  }
  // trace(Rn_inv @ Rs) + eps
  float trr = EPS5, tri = 0.f;
  for (int i = 0; i < 8; ++i)
    for (int j = 0; j < 8; ++j) {
      trr += B[i][j][0] * S[j][i][0] - B[i][j][1] * S[j][i][1];
      tri += B[i][j][0] * S[j][i][1] + B[i][j][1] * S[j][i][0];
    }
  const float td = 1.f / (trr * trr + tri * tri);

  float* wv = wv_out + (size_t)nf * 16;
  for (int i = 0; i < 8; ++i) {
    float nr = 0.f, ni = 0.f;
    for (int j = 0; j < 8; ++j) {
      nr += B[i][j][0] * sv[j][0] - B[i][j][1] * sv[j][1];
      ni += B[i][j][0] * sv[j][1] + B[i][j][1] * sv[j][0];
    }
    const float wr = (nr * trr + ni * tri) * td;
    const float wi = (ni * trr - nr * tri) * td;
    wv[i * 2] = wr;        // conj(w): (re, -im)
    wv[i * 2 + 1] = -wi;
  }
}

// ---------------------------------------------------------------------------
// Kernel 4: beamform + layout transpose. Tile (16 f) x (64 t) per block.
// Reads of x are t-contiguous (coalesced); writes go through LDS so each
// output row (n,t,f0..f0+15,{re,im}) is a contiguous 128B burst.
// ---------------------------------------------------------------------------
#define FT 16
#define TT 64
__global__ __launch_bounds__(256) void beam_kernel(const float* __restrict__ xr,
                                                   const float* __restrict__ xi,
                                                   const float* __restrict__ wv,
                                                   float* __restrict__ out) {
  const int n = blockIdx.z;
  const int f0 = blockIdx.y * FT;
  const int t0 = blockIdx.x * TT;
  const int tid = threadIdx.x;

  __shared__ float sw[FT][8][2];
  __shared__ float sm[FT][TT][2];
  {
    const int e = tid;  // FT*8*2 == 256
    const int fl = e >> 4, c = (e >> 1) & 7, comp = e & 1;
    const int f = f0 + fl;
    sw[fl][c][comp] =
        (f < F_DIM) ? wv[(((size_t)n * F_DIM + f) * 8 + c) * 2 + comp] : 0.f;
  }
  __syncthreads();

  const int tl = tid & 63;
  const int fg = tid >> 6;
  for (int fl = fg; fl < FT; fl += 4) {
    const int f = f0 + fl;
    const int t = t0 + tl;
    float accr = 0.f, acci = 0.f;
    if (f < F_DIM && t < T_DIM) {
#pragma unroll
      for (int c = 0; c < 8; ++c) {
        const size_t idx =
            (((size_t)n * C_DIM + c) * F_DIM + f) * (size_t)T_DIM + t;
        const float vr = sw[fl][c][0], vi = sw[fl][c][1];
        const float ar = xr[idx], ai = xi[idx];
        accr = fmaf(vr, ar, accr);
        accr = fmaf(-vi, ai, accr);
        acci = fmaf(vr, ai, acci);
        acci = fmaf(vi, ar, acci);
      }
    }
    sm[fl][tl][0] = accr;
    sm[fl][tl][1] = acci;
  }
  __syncthreads();

  const int wf = tid & 31;  // covers 32 floats = (16 f) x (re,im)
  const int wt = tid >> 5;
  for (int trow = wt; trow < TT; trow += 8) {
    const int t = t0 + trow;
    const int f = f0 + (wf >> 1);
    if (t < T_DIM && f < F_DIM)
      out[(((size_t)n * T_DIM + t) * F_DIM) * 2 + (size_t)f0 * 2 + wf] =
          sm[wf >> 1][trow][wf & 1];
  }
}

// ---------------------------------------------------------------------------
extern "C" void kernel_launch(void* const* d_in, const int* in_sizes, int n_in,
                              void* d_out, int out_size, void* d_ws,
                              size_t ws_size, hipStream_t stream) {
  const float* mask = (const float*)d_in[0];
  const float* xr = (const float*)d_in[1];
  const float* xi = (const float*)d_in[2];
  const float* proj_w = (const float*)d_in[3];
  const float* proj_b = (const float*)d_in[4];
  const float* gvec_w = (const float*)d_in[5];
  const float* gvec_b = (const float*)d_in[6];
  float* out = (float*)d_out;

  // workspace layout (floats): Rs | Rn | feat | g | conj(w)  (~2.3 MB total)
  float* ws = (float*)d_ws;
  float* Rs = ws;                                     // NF*128
  float* Rn = Rs + (size_t)NF * 128;                  // NF*128
  float* feat = Rn + (size_t)NF * 128;                // N*C*F
  float* g = feat + (size_t)N_DIM * C_DIM * F_DIM;    // N*C
  float* wv = g + N_DIM * C_DIM;                      // NF*16

  covar_kernel<<<NF, 128, 0, stream>>>(mask, xr, xi, Rs, Rn, feat);
  attn_kernel<<<N_DIM * C_DIM, 256, 0, stream>>>(feat, proj_w, proj_b, gvec_w,
                                                 gvec_b, g);
  solve_kernel<<<(NF + 63) / 64, 64, 0, stream>>>(Rs, Rn, g, wv);
  beam_kernel<<<dim3((T_DIM + TT - 1) / TT, (F_DIM + FT - 1) / FT, N_DIM), 256,
                0, stream>>>(xr, xi, wv, out);
}